// CRF_42949672961092
// MI455X (gfx1250) — compile-verified
//
#include <hip/hip_runtime.h>
#include <stdint.h>

// ---------------------------------------------------------------------------
// CRF Viterbi decode, gfx1250 (MI455X).
//   B=128 chains, L=1024 steps, T=130 states (START=128, END=129).
//   One workgroup per batch element; 288 threads = 9 wave32s.
//   Forward: thread (h,j) reduces i in [h*68, h*68+68) with trans column in
//            VGPRs, part[] ping-ponged in LDS (float4 ds reads), 2-way combine.
//   feats rows prefetched with GLOBAL_LOAD_ASYNC_TO_LDS_B32; backpointer
//   chunks staged with GLOBAL_LOAD_ASYNC_TO_LDS_B128 (ASYNCcnt +
//   s_wait_asynccnt). Backtrack fused: uint8 backpointer rows chunk-staged
//   into an LDS double buffer, serial pointer chase in LDS while the next
//   chunk streams in.
// ---------------------------------------------------------------------------

namespace {
constexpr int kB      = 128;
constexpr int kL      = 1024;
constexpr int kT      = 130;
constexpr int kStart  = 128;   // T-2
constexpr int kEnd    = 129;   // T-1
constexpr int kSeg    = 68;    // i-slice per segment (pad 130 -> 136)
constexpr int kNSeg   = 2;
constexpr int kJW     = 144;   // padded j slots per segment group
constexpr int kThreads = kNSeg * kJW;  // 288 threads = 9 waves
constexpr int kTP     = 136;   // padded part/feat length (16B aligned rows)
constexpr int kCT     = 128;   // backtrack chunk rows
constexpr int kBPS    = 132;   // backpointer row stride in bytes (4B aligned)
}

// ---- gfx1250 async global->LDS path (guarded; falls back to sync copies) ---
#if defined(__HIP_DEVICE_COMPILE__) && defined(__gfx1250__)
#if __has_builtin(__builtin_amdgcn_global_load_async_to_lds_b32) && \
    __has_builtin(__builtin_amdgcn_s_wait_asynccnt)
#define VIT_ASYNC 1
#endif
#endif
#ifndef VIT_ASYNC
#define VIT_ASYNC 0
#endif

#if VIT_ASYNC
typedef __attribute__((address_space(1))) int GInt;   // global (AS1)
typedef __attribute__((address_space(3))) int LInt;   // LDS    (AS3)
typedef int vit_v4i __attribute__((__vector_size__(4 * sizeof(int))));
typedef __attribute__((address_space(1))) vit_v4i GV4; // global v4i (AS1)
typedef __attribute__((address_space(3))) vit_v4i LV4; // LDS    v4i (AS3)
__device__ __forceinline__ void vit_copy_b32(const void* g, void* l) {
  __builtin_amdgcn_global_load_async_to_lds_b32(
      (GInt*)(unsigned long long)(g), (LInt*)(unsigned long long)(l), 0, 0);
}
#define VIT_WAIT_ASYNC(N) __builtin_amdgcn_s_wait_asynccnt(N)
#if __has_builtin(__builtin_amdgcn_global_load_async_to_lds_b128)
__device__ __forceinline__ void vit_copy_b128(const void* g, void* l) {
  __builtin_amdgcn_global_load_async_to_lds_b128(
      (GV4*)(unsigned long long)(g), (LV4*)(unsigned long long)(l), 0, 0);
}
#else
__device__ __forceinline__ void vit_copy_b128(const void* g, void* l) {
  vit_copy_b32((const char*)g + 0,  (char*)l + 0);
  vit_copy_b32((const char*)g + 4,  (char*)l + 4);
  vit_copy_b32((const char*)g + 8,  (char*)l + 8);
  vit_copy_b32((const char*)g + 12, (char*)l + 12);
}
#endif
#else
__device__ __forceinline__ void vit_copy_b32(const void* g, void* l) {
  *(unsigned int*)l = *(const unsigned int*)g;
}
__device__ __forceinline__ void vit_copy_b128(const void* g, void* l) {
  *(uint4*)l = *(const uint4*)g;
}
#define VIT_WAIT_ASYNC(N) ((void)0)
#endif

__device__ __forceinline__ int imin_(int a, int b) { return a < b ? a : b; }
__device__ __forceinline__ int imax_(int a, int b) { return a > b ? a : b; }

__global__ __launch_bounds__(kThreads) void viterbi_crf_kernel(
    const float* __restrict__ feats,          // (B, L, T) f32
    const unsigned char* __restrict__ mask,   // (B, L) bool (1 byte)
    const float* __restrict__ trans,          // (T, T) f32
    float* __restrict__ out,                  // (B, L) decoded states (as f32)
    unsigned char* __restrict__ bpG)          // ws: (B, L, kBPS) u8 backptrs
{
  __shared__ __attribute__((aligned(16))) float s_part[2][kTP];
  __shared__ __attribute__((aligned(16))) float s_feat[2][kTP];
  __shared__ float s_pm[kNSeg][kJW];
  __shared__ int   s_pi[kNSeg][kJW];
  __shared__ int   s_red[kThreads];
  __shared__ int   s_len;
  __shared__ int   s_ptr;
  __shared__ __attribute__((aligned(16))) unsigned char s_bt[2][kCT * kBPS];
  __shared__ int   s_dec[kCT];

  const int b   = blockIdx.x;
  const int tid = threadIdx.x;
  const int h   = tid / kJW;        // segment 0/1
  const int j   = tid % kJW;        // state (valid if < kT)
  const bool jA = (j < kT);

  const float*         fB  = feats + (size_t)b * kL * kT;
  const unsigned char* mB  = mask  + (size_t)b * kL;
  unsigned char*       bpB = bpG   + (size_t)b * kL * kBPS;

  // ---------------- sequence length = sum(mask[b]) ----------------
  int ps = 0;
  for (int t = tid; t < kL; t += kThreads) ps += (mB[t] ? 1 : 0);
  s_red[tid] = ps;
  // init padded part slots once (never overwritten)
  if (tid < 2 * (kTP - kT)) {
    const int bsel = tid / (kTP - kT);
    s_part[bsel][kT + tid % (kTP - kT)] = -1e30f;
  }
  __syncthreads();
  if (tid == 0) {
    int acc = 0;
    for (int k = 0; k < kThreads; ++k) acc += s_red[k];
    s_len = acc;
  }
  __syncthreads();
  const int len = s_len;            // in [256, 1024] for this problem

  // ------------- transition column slice in registers -------------
  float tc[kSeg];
  {
    const int i0 = h * kSeg;
#pragma unroll
    for (int k = 0; k < kSeg; ++k) {
      const int i = i0 + k;
      tc[k] = (jA && i < kT) ? trans[i * kT + j] : -1e30f;
    }
  }

  // ---------------- feat prefetch (double buffer) ----------------
  // lane tid<kT owns s_feat[buf][tid]; async completion is per-wave and the
  // consumer is the producing lane itself, so s_wait_asynccnt suffices.
  if (tid < kT) vit_copy_b32(fB + tid, &s_feat[0][tid]);
  if (len > 1 && tid < kT) vit_copy_b32(fB + (size_t)kT + tid, &s_feat[1][tid]);
  VIT_WAIT_ASYNC(0);
  __syncthreads();

  // part_0[j] = feat[0][j] + trans[START][j]
  if (tid < kT) s_part[0][tid] = s_feat[0][tid] + trans[kStart * kT + tid];
  __syncthreads();

  // ---------------------- forward recursion ----------------------
  for (int t = 1; t < len; ++t) {
    if (t + 1 < len && tid < kT)
      vit_copy_b32(fB + (size_t)(t + 1) * kT + tid, &s_feat[(t + 1) & 1][tid]);

    const int pb = (t - 1) & 1;     // part_{t-1} buffer
    float m = -3e38f;
    int   bi = 0;
    const float4* p4 = (const float4*)&s_part[pb][h * kSeg];
#pragma unroll
    for (int q = 0; q < kSeg / 4; ++q) {
      const float4 p = p4[q];
      float v;
      v = p.x + tc[4 * q + 0]; if (v > m) { m = v; bi = 4 * q + 0; }
      v = p.y + tc[4 * q + 1]; if (v > m) { m = v; bi = 4 * q + 1; }
      v = p.z + tc[4 * q + 2]; if (v > m) { m = v; bi = 4 * q + 2; }
      v = p.w + tc[4 * q + 3]; if (v > m) { m = v; bi = 4 * q + 3; }
    }
    s_pm[h][j] = m;
    s_pi[h][j] = bi;
    // ensure this step's feat row landed (newest prefetch may stay in flight)
    if (t + 1 < len) { VIT_WAIT_ASYNC(1); } else { VIT_WAIT_ASYNC(0); }
    __syncthreads();

    if (h == 0 && jA) {
      const float m0 = s_pm[0][j];
      const float m1 = s_pm[1][j];
      float mm; int ii;
      if (m1 > m0) { mm = m1; ii = kSeg + s_pi[1][j]; }   // strict >: low i wins ties
      else         { mm = m0; ii = s_pi[0][j]; }
      s_part[pb ^ 1][j] = mm + s_feat[t & 1][j];
      // mask[b][t] is true for all t < len (monotone mask), so bp unmasked here
      bpB[(size_t)t * kBPS + j] = (unsigned char)ii;
    }
    __syncthreads();
  }

  VIT_WAIT_ASYNC(0);
  __threadfence();                  // make bp stores visible for re-read below
  __syncthreads();

  // -------- final pointer P = argmax_i(part_{len-1}[i] + trans[i][END]) ----
  const int pbuf = (len - 1) & 1;
  if (tid < kT) s_pm[0][tid] = s_part[pbuf][tid] + trans[tid * kT + kEnd];
  __syncthreads();
  if (tid == 0) {
    float m = s_pm[0][0];
    int bi = 0;
    for (int i = 1; i < kT; ++i) {
      const float v = s_pm[0][i];
      if (v > m) { m = v; bi = i; }   // first-occurrence argmax
    }
    s_ptr = bi;
  }
  __syncthreads();
  const int P = s_ptr;

  // -------- tail writes: decode[L-1]=P, decode[len-1]=P, zeros between -----
  for (int t = tid; t < kL; t += kThreads) {
    if (t == kL - 1 || t == len - 1)      out[(size_t)b * kL + t] = (float)P;
    else if (t >= len)                    out[(size_t)b * kL + t] = 0.0f;
  }

  // ---------------- backtrack with chunked LDS staging ----------------
  const int ccHi   = (len - 1) / kCT;
  const int xfers = kCT * kBPS / 16;   // 16-byte transfers per chunk (1056)

  // load chunk ccHi into buffer 0
  {
    const unsigned char* src = bpB + (size_t)ccHi * kCT * kBPS;
    for (int w = tid; w < xfers; w += kThreads)
      vit_copy_b128(src + 16 * w, &s_bt[0][16 * w]);
  }
  VIT_WAIT_ASYNC(0);
  __syncthreads();

  int bufi = 0;
  int ptr  = P;
  for (int cc = ccHi; cc >= 0; --cc) {
    if (cc > 0) {       // prefetch next chunk while chasing this one
      const unsigned char* src = bpB + (size_t)(cc - 1) * kCT * kBPS;
      for (int w = tid; w < xfers; w += kThreads)
        vit_copy_b128(src + 16 * w, &s_bt[bufi ^ 1][16 * w]);
    }
    const int c0  = cc * kCT;
    const int sHi = imin_(len - 1, c0 + kCT - 1);
    const int sLo = imax_(1, c0);
    if (tid == 0) {
      int p2 = ptr;
      const unsigned char* rows = s_bt[bufi];
      for (int s = sHi; s >= sLo; --s) {      // decode[s-1] = bp_row[s][p2]
        p2 = rows[(s - c0) * kBPS + p2];
        s_dec[s - c0] = p2;
      }
      s_ptr = p2;
    }
    __syncthreads();
    ptr = s_ptr;
    for (int s = sLo + tid; s <= sHi; s += kThreads)
      out[(size_t)b * kL + (s - 1)] = (float)s_dec[s - c0];
    if (cc > 0) {
      VIT_WAIT_ASYNC(0);
      __syncthreads();
    }
    bufi ^= 1;
  }
}

extern "C" void kernel_launch(void* const* d_in, const int* in_sizes, int n_in,
                              void* d_out, int out_size, void* d_ws, size_t ws_size,
                              hipStream_t stream) {
  (void)in_sizes; (void)n_in; (void)out_size; (void)ws_size;
  const float*         feats = (const float*)d_in[0];         // (B,L,T) f32
  const unsigned char* mask  = (const unsigned char*)d_in[1]; // (B,L) bool
  const float*         trans = (const float*)d_in[2];         // (T,T) f32
  float*               out   = (float*)d_out;                 // (B,L) states
  unsigned char*       bp    = (unsigned char*)d_ws;          // B*L*kBPS bytes

  viterbi_crf_kernel<<<dim3(kB), dim3(kThreads), 0, stream>>>(
      feats, mask, trans, out, bp);
}